// CurveGraphic2d_62216896250461
// MI455X (gfx1250) — compile-verified
//
#include <hip/hip_runtime.h>
#include <math.h>

#define NSAMP 32

typedef float v2f __attribute__((ext_vector_type(2)));
typedef float v8f __attribute__((ext_vector_type(8)));

// ---------------------------------------------------------------------------
// Phase 1: arc-length-uniform Bezier sample points. 16 curves, 1 thread each.
// Writes sp[b][s][2] (c0 pairs with y, c1 pairs with x) into workspace.
// ---------------------------------------------------------------------------
__global__ void bezier_sample_kernel(const float* __restrict__ inputs,
                                     float* __restrict__ sp) {
  int b = threadIdx.x;
  if (b >= 16) return;
  float k0[4], k1[4];
#pragma unroll
  for (int k = 0; k < 4; ++k) {
    k0[k] = inputs[b * 8 + k * 2 + 0] * 512.0f;  // canvas H
    k1[k] = inputs[b * 8 + k * 2 + 1] * 512.0f;  // canvas W
  }
  float d0[NSAMP], d1[NSAMP], cum[NSAMP];
  for (int s = 0; s < NSAMP; ++s) {
    float t = (float)s * (1.0f / 31.0f);
    float u = 1.0f - t;
    float b0 = u * u * u;
    float b1 = 3.0f * t * u * u;
    float b2 = 3.0f * t * t * u;
    float b3 = t * t * t;
    d0[s] = b0 * k0[0] + b1 * k0[1] + b2 * k0[2] + b3 * k0[3];
    d1[s] = b0 * k1[0] + b1 * k1[1] + b2 * k1[2] + b3 * k1[3];
  }
  cum[0] = 0.0f;
  for (int s = 1; s < NSAMP; ++s) {
    float dy = d0[s] - d0[s - 1];
    float dx = d1[s] - d1[s - 1];
    cum[s] = cum[s - 1] + sqrtf(dy * dy + dx * dx);
  }
  float total = cum[NSAMP - 1];
  for (int s = 0; s < NSAMP; ++s) {
    float target = (float)s * (1.0f / 31.0f) * total;
    int j = 0;
    for (int i = 1; i < NSAMP - 1; ++i)
      if (cum[i] <= target) j = i;
    float den = cum[j + 1] - cum[j];
    float frac = (den > 0.0f) ? (target - cum[j]) / den : 0.0f;
    sp[b * (NSAMP * 2) + s * 2 + 0] = d0[j] + frac * (d0[j + 1] - d0[j]);
    sp[b * (NSAMP * 2) + s * 2 + 1] = d1[j] + frac * (d1[j + 1] - d1[j]);
  }
}

// ---------------------------------------------------------------------------
// Phase 2: distance field + shading via V_WMMA_F32_16X16X4_F32.
// Samples on M, pixels on N:
//   A[m=sample] = [-2*sy, -2*sx, 1, s^2],  B[:,n=pixel] = [y, x, p^2, 1]
//   => D[m][n] = squared distance, exact.
// One wave = 32 consecutive pixels (two 16-wide N tiles). Per curve:
//   4 WMMAs, per-lane min over 16 D registers, one ds_swizzle SWAPX16 to
//   merge the partner half-wave's 16 samples, then a divergence-free
//   sqrt/pow epilogue with one coalesced store per lane.
// ---------------------------------------------------------------------------
#define SWIZ_SWAPX16 0x401F  // group-of-32: xor=0x10, or=0x00, and=0x1f

__global__ __launch_bounds__(256) void curve_raster_kernel(
    const float* __restrict__ sp, const float* __restrict__ widths,
    const float* __restrict__ aafac, float* __restrict__ out) {
  const int lane = threadIdx.x & 31;
  const int wave = threadIdx.x >> 5;
  const int tile = blockIdx.x * 8 + wave;  // 32-pixel tile id, [0, 8192)
  const int n0 = tile * 32;                // linear pixel base (row-aligned)
  const int y = n0 >> 9;
  const int xbase = n0 & 511;
  const float fy = (float)y;

  const int lm = lane & 15;
  const bool hi = lane >= 16;

  // B-matrix 4x16 f32 (K x N-pixels), loop-invariant.
  // lanes 0-15: N=lane, K=0 in v0 (=y), K=1 in v1 (=x)
  // lanes 16-31: N=lane-16, K=2 in v0 (=p^2), K=3 in v1 (=1)
  v2f Blo, Bhi;
  {
    float xl = (float)(xbase + lm);         // pixel lm of lo tile
    float xh = (float)(xbase + 16 + lm);    // pixel lm of hi tile
    if (!hi) {
      Blo.x = fy;  Blo.y = xl;
      Bhi.x = fy;  Bhi.y = xh;
    } else {
      Blo.x = fy * fy + xl * xl;  Blo.y = 1.0f;
      Bhi.x = fy * fy + xh * xh;  Bhi.y = 1.0f;
    }
  }

  for (int b = 0; b < 16; ++b) {
    const float* spb = sp + b * (NSAMP * 2);
    // sample lm (A0 rows) and sample 16+lm (A1 rows)
    float c0a = spb[lm * 2 + 0];
    float c1a = spb[lm * 2 + 1];
    float c0b = spb[(16 + lm) * 2 + 0];
    float c1b = spb[(16 + lm) * 2 + 1];

    // A-matrix 16x4 f32 (M-samples x K):
    // lanes 0-15: M=lane, K=0/1 in v0/v1; lanes 16-31: M=lane-16, K=2/3.
    v2f A0, A1;
    if (!hi) {
      A0.x = -2.0f * c0a;  A0.y = -2.0f * c1a;
      A1.x = -2.0f * c0b;  A1.y = -2.0f * c1b;
    } else {
      A0.x = 1.0f;  A0.y = c0a * c0a + c1a * c1a;
      A1.x = 1.0f;  A1.y = c0b * c0b + c1b * c1b;
    }

    v8f cz = {0.f, 0.f, 0.f, 0.f, 0.f, 0.f, 0.f, 0.f};
    v8f d00 = __builtin_amdgcn_wmma_f32_16x16x4_f32(
        false, A0, false, Blo, (short)0, cz, false, false);
    v8f d01 = __builtin_amdgcn_wmma_f32_16x16x4_f32(
        false, A1, false, Blo, (short)0, cz, false, false);
    v8f d10 = __builtin_amdgcn_wmma_f32_16x16x4_f32(
        false, A0, false, Bhi, (short)0, cz, false, false);
    v8f d11 = __builtin_amdgcn_wmma_f32_16x16x4_f32(
        false, A1, false, Bhi, (short)0, cz, false, false);

    // Per-lane min over the 16 sample-rows this lane holds, per pixel tile.
    float mlo = fminf(d00[0], d01[0]);
    float mhi = fminf(d10[0], d11[0]);
#pragma unroll
    for (int i = 1; i < 8; ++i) {
      mlo = fminf(mlo, fminf(d00[i], d01[i]));
      mhi = fminf(mhi, fminf(d10[i], d11[i]));
    }
    // Merge partner half-wave (other 16 samples of the same pixel).
    float plo = __int_as_float(
        __builtin_amdgcn_ds_swizzle(__float_as_int(mlo), SWIZ_SWAPX16));
    float phi = __int_as_float(
        __builtin_amdgcn_ds_swizzle(__float_as_int(mhi), SWIZ_SWAPX16));
    mlo = fminf(mlo, plo);
    mhi = fminf(mhi, phi);

    // lane L<16 -> pixel n0+L from lo tile; lane L>=16 -> pixel n0+L from hi.
    float d2min = hi ? mhi : mlo;
    float dist = sqrtf(fmaxf(d2min, 1e-12f));
    float w = widths[b];
    float a = aafac[b];
    float val = (dist < w) ? (1.0f - powf(dist / w + 1e-6f, a)) : 0.0f;
    out[b * (512 * 512) + n0 + lane] = val;
  }
}

// ---------------------------------------------------------------------------
extern "C" void kernel_launch(void* const* d_in, const int* in_sizes, int n_in,
                              void* d_out, int out_size, void* d_ws,
                              size_t ws_size, hipStream_t stream) {
  const float* inputs = (const float*)d_in[0];  // [16,4,2]
  const float* widths = (const float*)d_in[1];  // [16]
  const float* aafac = (const float*)d_in[2];   // [16]
  float* out = (float*)d_out;                   // [16,512,512]
  float* sp = (float*)d_ws;                     // [16,32,2] sample points

  bezier_sample_kernel<<<1, 32, 0, stream>>>(inputs, sp);
  // 262144 pixels / 32 per wave = 8192 waves; 8 waves per 256-thread block
  curve_raster_kernel<<<8192 / 8, 256, 0, stream>>>(sp, widths, aafac, out);
}